// SGNHeadOED_58806692216890
// MI455X (gfx1250) — compile-verified
//
#include <hip/hip_runtime.h>
#include <cstdint>

// CSPN 3-axis affinity propagation, prop_time=1.
//   guidance: (24, 128, 128, 16) f32   d_in[0]
//   blur    : (20, 128, 128, 16) f32   d_in[1]
//   out     : (20, 128, 128, 16) f32   d_out
// Three sequential 8-neighbor gated stencil passes:
//   pass0 in (h,w), pass1 in (h,z), pass2 in (w,z); gates from guidance planes
//   [8p..8p+8), shifted per neighbor, abs-sum normalized, shared over channels.

#define HD 128
#define WD 128
#define ZD 16
#define CCH 20
#define PLANE (HD * WD * ZD)   // 262144 spatial sites
#define TPB 256

// Generic shared-memory pointer -> raw LDS byte offset (low 32 bits of the
// flat LDS aperture address hold the wave-relative LDS offset).
__device__ __forceinline__ unsigned lds_byte_addr(const void* p) {
  return (unsigned)(uintptr_t)p;
}

// Per-channel body, fully unrolled so s_wait_asynccnt gets a literal immediate.
// Retires async center loads in issue order: channel C needs asynccnt <= CCH-1-C.
template <int C>
struct ChanLoop {
  static __device__ __forceinline__ void run(const float* __restrict__ in,
                                             float* __restrict__ out, int n, int tid,
                                             const float* ctr_s, const float (&wgt)[8],
                                             const int (&nbr)[8], float onems) {
    asm volatile("s_wait_asynccnt %0" ::"n"(CCH - 1 - C) : "memory");
    const float* bc = in + (size_t)C * PLANE + n;   // center ptr, this channel
    float acc = onems * ctr_s[C * TPB + tid];       // center value from LDS (async-staged)
    #pragma unroll
    for (int k = 0; k < 8; ++k)
      acc = fmaf(wgt[k], bc[nbr[k]], acc);          // nbr clamped to 0 offset when wgt==0
    out[(size_t)C * PLANE + n] = acc;
    ChanLoop<C + 1>::run(in, out, n, tid, ctr_s, wgt, nbr, onems);
  }
};
template <>
struct ChanLoop<CCH> {
  static __device__ __forceinline__ void run(const float*, float*, int, int,
                                             const float*, const float (&)[8],
                                             const int (&)[8], float) {}
};

template <int PASS>
__global__ __launch_bounds__(TPB) void cspn_pass_kernel(const float* __restrict__ guid,
                                                        const float* __restrict__ in,
                                                        float* __restrict__ out) {
  __shared__ float ctr_s[CCH * TPB];   // 20 KB: per-thread center value, 20 channels

  const int tid = threadIdx.x;
  const int n = blockIdx.x * TPB + tid;     // flat (h,w,z); grid covers PLANE exactly
  const int z = n & (ZD - 1);
  const int w = (n >> 4) & (WD - 1);
  const int h = n >> 11;

  // ---- Issue 20 async center loads (global -> LDS), overlap with weight math ----
  const unsigned lbase = lds_byte_addr(&ctr_s[0]) + (unsigned)tid * 4u;
  #pragma unroll
  for (int c = 0; c < CCH; ++c) {
    unsigned long long ga =
        (unsigned long long)(uintptr_t)(in + (size_t)c * PLANE + n);
    unsigned la = lbase + (unsigned)(c * TPB * 4);
    asm volatile("global_load_async_to_lds_b32 %0, %1, off" ::"v"(la), "v"(ga)
                 : "memory");
  }

  // ---- Pass geometry: spatial dims (i,j) with strides (s0,s1) in the flat layout ----
  int i, j, N0, N1, s0, s1;
  if constexpr (PASS == 0) { i = h; j = w; N0 = HD; N1 = WD; s0 = WD * ZD; s1 = ZD; }
  else if constexpr (PASS == 1) { i = h; j = z; N0 = HD; N1 = ZD; s0 = WD * ZD; s1 = 1; }
  else { i = w; j = z; N0 = WD; N1 = ZD; s0 = ZD; s1 = 1; }

  // Neighbor offsets per gate k (derived from the ZeroPad2d patterns).
  constexpr int DI[8] = {1, 1, 1, 0, 0, -1, -1, -1};
  constexpr int DJ[8] = {1, 0, -1, 1, -1, 1, 0, -1};

  const float* gp = guid + (size_t)PASS * 8 * PLANE + n;

  // ---- Weights: w_k = guidance[8p+k] at shifted site, abs-sum normalized ----
  float wgt[8];
  int nbr[8];
  float norm = 0.0f;
  #pragma unroll
  for (int k = 0; k < 8; ++k) {
    const bool ok = ((unsigned)(i + DI[k]) < (unsigned)N0) &&
                    ((unsigned)(j + DJ[k]) < (unsigned)N1);
    const int off = DI[k] * s0 + DJ[k] * s1;
    const float g = ok ? gp[(size_t)k * PLANE + off] : 0.0f;
    wgt[k] = g;
    nbr[k] = ok ? off : 0;   // clamped: contribution is zeroed by wgt[k]==0
    norm += fabsf(g);
  }
  const float inv = 1.0f / norm;   // norm==0 -> NaN, matching reference 0/0
  float gsum = 0.0f;
  #pragma unroll
  for (int k = 0; k < 8; ++k) { wgt[k] *= inv; gsum += wgt[k]; }
  const float onems = 1.0f - gsum;

  // ---- Apply to all 20 channels, retiring async loads in order ----
  ChanLoop<0>::run(in, out, n, tid, ctr_s, wgt, nbr, onems);
}

extern "C" void kernel_launch(void* const* d_in, const int* in_sizes, int n_in,
                              void* d_out, int out_size, void* d_ws, size_t ws_size,
                              hipStream_t stream) {
  (void)in_sizes; (void)n_in; (void)out_size; (void)ws_size;
  const float* guid = (const float*)d_in[0];   // (24,128,128,16)
  const float* blur = (const float*)d_in[1];   // (20,128,128,16)
  float* out = (float*)d_out;                  // (20,128,128,16)
  float* tmp = (float*)d_ws;                   // needs PLANE*CCH*4 = 20 MiB scratch

  dim3 grid(PLANE / TPB), block(TPB);
  // x-pass: blur -> out ; y-pass: out -> tmp ; z-pass: tmp -> out
  cspn_pass_kernel<0><<<grid, block, 0, stream>>>(guid, blur, out);
  cspn_pass_kernel<1><<<grid, block, 0, stream>>>(guid, out, tmp);
  cspn_pass_kernel<2><<<grid, block, 0, stream>>>(guid, tmp, out);
}